// HeteroGNN_2035814499086
// MI455X (gfx1250) — compile-verified
//
#include <hip/hip_runtime.h>
#include <hip/hip_bf16.h>

typedef __attribute__((ext_vector_type(2))) float v2f;
typedef __attribute__((ext_vector_type(8))) float v8f;

#define FDIM 128
#define EPS  1e-5f

// -----------------------------------------------------------------------------
// GEMM: Y[N,128] = X[N,128] @ W[128,128] using V_WMMA_F32_16X16X4_F32.
// Block = 256 threads (8 waves). Block tile: 64 rows x 128 cols.
// Wave w owns cols [16w, 16w+16) across all 64 rows: 4 x (16x16) accumulators.
// One B fragment (2 global b32 loads) feeds 4 WMMAs -> 4x less W traffic than
// a 16-row tile. A tile (64x128 = 32 KB) staged in LDS via coalesced float4.
// -----------------------------------------------------------------------------
__global__ __launch_bounds__(256) void gemm_f32_wmma(
    const float* __restrict__ X, const float* __restrict__ W,
    float* __restrict__ Y, int Nrows)
{
    __shared__ float As[64 * FDIM];          // 32 KB

    const int t       = threadIdx.x;
    const int rowTile = blockIdx.x * 64;

    // Coalesced cooperative load of the 64x128 A tile (contiguous row-major):
    // 2048 float4s across 256 threads = 8 each.
    {
        const float4* __restrict__ Xv = (const float4*)(X + (size_t)rowTile * FDIM);
        float4* Asv = (float4*)As;
        #pragma unroll
        for (int i = 0; i < 8; ++i)
            Asv[t + i * 256] = Xv[t + i * 256];
    }
    __syncthreads();

    const int wave = t >> 5;
    const int lane = t & 31;
    const int half = lane >> 4;   // 0: K pair {k,k+1}; 1: {k+2,k+3}
    const int m    = lane & 15;
    const int c0   = wave * 16;

    v8f acc0 = {}, acc1 = {}, acc2 = {}, acc3 = {};

    #pragma unroll
    for (int k0 = 0; k0 < FDIM; k0 += 4) {
        const int ka = k0 + 2 * half;
        // B fragment: 4x16, rows ka/ka+1, col c0+m  (shared by 4 row sub-tiles)
        v2f b;
        b.x = W[(size_t)ka       * FDIM + c0 + m];
        b.y = W[(size_t)(ka + 1) * FDIM + c0 + m];

        // A fragments: row sub-tile r covers rows [16r, 16r+16)
        v2f a0, a1, a2, a3;
        a0.x = As[(0 * 16 + m) * FDIM + ka];  a0.y = As[(0 * 16 + m) * FDIM + ka + 1];
        a1.x = As[(1 * 16 + m) * FDIM + ka];  a1.y = As[(1 * 16 + m) * FDIM + ka + 1];
        a2.x = As[(2 * 16 + m) * FDIM + ka];  a2.y = As[(2 * 16 + m) * FDIM + ka + 1];
        a3.x = As[(3 * 16 + m) * FDIM + ka];  a3.y = As[(3 * 16 + m) * FDIM + ka + 1];

        // 8 args: (neg_a, A, neg_b, B, c_mod, C, reuse_a, reuse_b)
        acc0 = __builtin_amdgcn_wmma_f32_16x16x4_f32(false, a0, false, b, (short)0, acc0, false, false);
        acc1 = __builtin_amdgcn_wmma_f32_16x16x4_f32(false, a1, false, b, (short)0, acc1, false, false);
        acc2 = __builtin_amdgcn_wmma_f32_16x16x4_f32(false, a2, false, b, (short)0, acc2, false, false);
        acc3 = __builtin_amdgcn_wmma_f32_16x16x4_f32(false, a3, false, b, (short)0, acc3, false, false);
    }

    // C/D layout: VGPR v -> row (v + 8*half), col m
    #pragma unroll
    for (int v = 0; v < 8; ++v) {
        const int rr = v + 8 * half;
        Y[(size_t)(rowTile +  0 + rr) * FDIM + c0 + m] = acc0[v];
        Y[(size_t)(rowTile + 16 + rr) * FDIM + c0 + m] = acc1[v];
        Y[(size_t)(rowTile + 32 + rr) * FDIM + c0 + m] = acc2[v];
        Y[(size_t)(rowTile + 48 + rr) * FDIM + c0 + m] = acc3[v];
    }
}

// -----------------------------------------------------------------------------
// Edge scatter: one wave per edge. Lane l loads float4 msg[src][4l..4l+3],
// scales by ew, atomically accumulates into acc[dst]. Relaxed agent-scope
// f32 atomics -> global_atomic_add_f32 in L2.
// -----------------------------------------------------------------------------
__global__ __launch_bounds__(256) void edge_scatter(
    const float* __restrict__ msg, const int* __restrict__ src,
    const int* __restrict__ dst, const float* __restrict__ ew,
    float* __restrict__ acc, int E)
{
    const int wave = threadIdx.x >> 5;
    const int lane = threadIdx.x & 31;
    const int e    = blockIdx.x * 8 + wave;
    if (e >= E) return;

    const int   s = src[e];
    const int   d = dst[e];
    const float w = ew[e];

    const float4 v = ((const float4*)(msg + (size_t)s * FDIM))[lane];
    float* a = acc + (size_t)d * FDIM + lane * 4;
    __hip_atomic_fetch_add(a + 0, v.x * w, __ATOMIC_RELAXED, __HIP_MEMORY_SCOPE_AGENT);
    __hip_atomic_fetch_add(a + 1, v.y * w, __ATOMIC_RELAXED, __HIP_MEMORY_SCOPE_AGENT);
    __hip_atomic_fetch_add(a + 2, v.z * w, __ATOMIC_RELAXED, __HIP_MEMORY_SCOPE_AGENT);
    __hip_atomic_fetch_add(a + 3, v.w * w, __ATOMIC_RELAXED, __HIP_MEMORY_SCOPE_AGENT);
}

// -----------------------------------------------------------------------------
// acc[n][f] = b0[f] + b1[f] + b2[f] + b3[f]
// -----------------------------------------------------------------------------
__global__ __launch_bounds__(256) void bias_init(
    float* __restrict__ acc, const float* __restrict__ b0,
    const float* __restrict__ b1, const float* __restrict__ b2,
    const float* __restrict__ b3, size_t total)
{
    size_t i = (size_t)blockIdx.x * 256 + threadIdx.x;
    if (i >= total) return;
    int f = (int)(i & (FDIM - 1));
    acc[i] = b0[f] + b1[f] + b2[f] + b3[f];
}

// -----------------------------------------------------------------------------
// LayerNorm over 128 features: one wave per row, 4 elems/lane, wave32
// __shfl_xor butterfly reduction.
// -----------------------------------------------------------------------------
__global__ __launch_bounds__(256) void layernorm_row(
    const float* __restrict__ in, const float* __restrict__ g,
    const float* __restrict__ b, float* __restrict__ out, int Nrows)
{
    const int wave = threadIdx.x >> 5;
    const int lane = threadIdx.x & 31;
    const int row  = blockIdx.x * 8 + wave;
    if (row >= Nrows) return;

    const float4 v = ((const float4*)(in + (size_t)row * FDIM))[lane];
    float s  = v.x + v.y + v.z + v.w;
    float sq = v.x * v.x + v.y * v.y + v.z * v.z + v.w * v.w;
    #pragma unroll
    for (int m = 16; m >= 1; m >>= 1) {
        s  += __shfl_xor(s,  m, 32);
        sq += __shfl_xor(sq, m, 32);
    }
    const float mu  = s * (1.0f / FDIM);
    const float var = sq * (1.0f / FDIM) - mu * mu;
    const float rs  = rsqrtf(var + EPS);

    const int f = lane * 4;
    float4 o;
    o.x = (v.x - mu) * rs * g[f + 0] + b[f + 0];
    o.y = (v.y - mu) * rs * g[f + 1] + b[f + 1];
    o.z = (v.z - mu) * rs * g[f + 2] + b[f + 2];
    o.w = (v.w - mu) * rs * g[f + 3] + b[f + 3];
    ((float4*)(out + (size_t)row * FDIM))[lane] = o;
}

// -----------------------------------------------------------------------------
// Input dict order: x, then per type t in {fwd,ons,sus,rest}:
//   ei_t [2,E] i32, ew_t [E] f32, W1_t [128,128], b1_t [128], W2_t, b2_t,
// then g1, be1, g2, be2.
// -----------------------------------------------------------------------------
extern "C" void kernel_launch(void* const* d_in, const int* in_sizes, int n_in,
                              void* d_out, int out_size, void* d_ws, size_t ws_size,
                              hipStream_t stream)
{
    const float* x = (const float*)d_in[0];
    const int*   ei[4];
    const float *ew[4], *W1[4], *b1[4], *W2[4], *b2[4];
    for (int t = 0; t < 4; ++t) {
        int base = 1 + t * 6;
        ei[t] = (const int*)d_in[base + 0];
        ew[t] = (const float*)d_in[base + 1];
        W1[t] = (const float*)d_in[base + 2];
        b1[t] = (const float*)d_in[base + 3];
        W2[t] = (const float*)d_in[base + 4];
        b2[t] = (const float*)d_in[base + 5];
    }
    const float* g1  = (const float*)d_in[25];
    const float* be1 = (const float*)d_in[26];
    const float* g2  = (const float*)d_in[27];
    const float* be2 = (const float*)d_in[28];

    const int E = in_sizes[2];            // ew_fwd element count
    const int N = in_sizes[0] / FDIM;     // rows of x
    const size_t total = (size_t)N * FDIM;

    float* out = (float*)d_out;
    float* tmp = (float*)d_ws;            // [N,128] per-type GEMM result
    float* acc = tmp + total;             // [N,128] scatter accumulator
    float* h   = out;                     // layer-1 output lives in d_out

    const int gBias = (int)((total + 255) / 256);
    const int gGemm = N / 64;             // 64-row block tiles
    const int gScat = (E + 7) / 8;
    const int gLN   = (N + 7) / 8;

    // ---- Layer 1 ----
    bias_init<<<gBias, 256, 0, stream>>>(acc, b1[0], b1[1], b1[2], b1[3], total);
    for (int t = 0; t < 4; ++t) {
        gemm_f32_wmma<<<gGemm, 256, 0, stream>>>(x, W1[t], tmp, N);
        edge_scatter<<<gScat, 256, 0, stream>>>(tmp, ei[t], ei[t] + E, ew[t], acc, E);
    }
    layernorm_row<<<gLN, 256, 0, stream>>>(acc, g1, be1, h, N);

    // ---- Layer 2 ----
    bias_init<<<gBias, 256, 0, stream>>>(acc, b2[0], b2[1], b2[2], b2[3], total);
    for (int t = 0; t < 4; ++t) {
        gemm_f32_wmma<<<gGemm, 256, 0, stream>>>(h, W2[t], tmp, N);
        edge_scatter<<<gScat, 256, 0, stream>>>(tmp, ei[t], ei[t] + E, ew[t], acc, E);
    }
    layernorm_row<<<gLN, 256, 0, stream>>>(acc, g2, be2, out, N);
}